// FeatureSelectionModule_76544907149592
// MI455X (gfx1250) — compile-verified
//
#include <hip/hip_runtime.h>
#include <hip/hip_bf16.h>

typedef __attribute__((ext_vector_type(2))) float v2f;
typedef __attribute__((ext_vector_type(8))) float v8f;

#define NPTS   8192
#define KNBR   32
#define NTOPK  819      // int(0.8 * 1024)
#define NSAMP  15
#define NCH    512
#define PANEL  512
#define SDS    520      // padded panel row stride (floats)

// ---------------------------------------------------------------------------
// helpers
// ---------------------------------------------------------------------------
__device__ __forceinline__ unsigned int fmono(float f) {
  // order-preserving float -> uint (ascending)
  unsigned int b = __float_as_uint(f);
  return b ^ ((b & 0x80000000u) ? 0xFFFFFFFFu : 0x80000000u);
}

__device__ __forceinline__ unsigned long long wave_min_u64(unsigned long long v) {
#pragma unroll
  for (int off = 16; off > 0; off >>= 1) {
    unsigned long long o = __shfl_xor(v, off, 32);
    v = (o < v) ? o : v;
  }
  return v;
}

__device__ __forceinline__ float wave_sum_f32(float v) {
#pragma unroll
  for (int off = 16; off > 0; off >>= 1) v += __shfl_xor(v, off, 32);
  return v;
}

__device__ __forceinline__ float mineig3(float a00, float a01, float a02,
                                         float a11, float a12, float a22) {
  float p1 = a01 * a01 + a02 * a02 + a12 * a12;
  float q  = (a00 + a11 + a22) * (1.0f / 3.0f);
  float b00 = a00 - q, b11 = a11 - q, b22 = a22 - q;
  float p2 = b00 * b00 + b11 * b11 + b22 * b22 + 2.0f * p1;
  if (p2 <= 1e-30f) return q;
  float p   = sqrtf(p2 * (1.0f / 6.0f));
  float inv = 1.0f / p;
  float c00 = b00 * inv, c01 = a01 * inv, c02 = a02 * inv;
  float c11 = b11 * inv, c12 = a12 * inv, c22 = a22 * inv;
  float detB = c00 * (c11 * c22 - c12 * c12)
             - c01 * (c01 * c22 - c12 * c02)
             + c02 * (c01 * c12 - c11 * c02);
  float r = 0.5f * detB;
  r = fminf(1.0f, fmaxf(-1.0f, r));
  float phi = acosf(r) * (1.0f / 3.0f);
  // smallest eigenvalue: q + 2p*cos(phi + 2*pi/3)
  return q + 2.0f * p * cosf(phi + 2.0943951023931953f);
}

// ---------------------------------------------------------------------------
// K1: Harris min-eigenvalue per point.  One workgroup per 16-row tile.
//     Distances via V_WMMA_F32_16X16X4_F32, streaming top-32 in LDS.
// ---------------------------------------------------------------------------
extern "C" __global__ void __launch_bounds__(256)
harris16x16x4(const float* __restrict__ src, const float* __restrict__ tgt,
              float* __restrict__ hout /* 2*NPTS */) {
  extern __shared__ char smem_h[];
  float*              sd   = (float*)smem_h;                          // 16*SDS
  unsigned long long* cand = (unsigned long long*)(sd + 16 * SDS);    // 16*KNBR*32
  int*                midx = (int*)(cand + 16 * KNBR * 32);           // 16*32

  const int cloud = blockIdx.x >> 9;        // 512 tiles per cloud
  const int tile  = blockIdx.x & 511;
  const float* X = (cloud ? tgt : src);
  const float* Y = X + NPTS;
  const float* Z = X + 2 * NPTS;

  const int t    = threadIdx.x;
  const int lane = t & 31;
  const int wave = t >> 5;
  const int half = lane >> 4;   // 0: K=0,1  1: K=2,3 (pad)
  const int l15  = lane & 15;

  // init candidate lists to +inf keys
  for (int i = t; i < 16 * KNBR * 32; i += 256) cand[i] = ~0ull;

  // A matrix (16x4 f32): lane l15 = row M, halves carry K={0,1}/{2,3}
  const int row0 = tile * 16;
  float xr = X[row0 + l15], yr = Y[row0 + l15], zr = Z[row0 + l15];
  v2f A;
  A.x = half ? zr : xr;
  A.y = half ? 0.0f : yr;

  // per-lane row norm -> broadcast row norms for the 8 D-rows this lane owns
  float aa_lane = xr * xr + yr * yr + zr * zr;   // valid in lanes 0..15
  float aam[8];
#pragma unroll
  for (int r = 0; r < 8; ++r) aam[r] = __shfl(aa_lane, r + 8 * half, 32);

  __syncthreads();

  unsigned long long curmax0 = ~0ull, curmax1 = ~0ull;

  for (int pp = 0; pp < NPTS / PANEL; ++pp) {
    const int j0 = pp * PANEL;
    // ---- distance panel via WMMA: 32 subtiles of 16 cols, 4 per wave ----
#pragma unroll
    for (int q = 0; q < 4; ++q) {
      const int st = wave + 8 * q;
      const int c0 = j0 + st * 16;
      float xc = X[c0 + l15], yc = Y[c0 + l15], zc = Z[c0 + l15];
      float bb = xc * xc + yc * yc + zc * zc;
      v2f B;
      B.x = half ? zc : xc;
      B.y = half ? 0.0f : yc;
      v8f acc = {0.f, 0.f, 0.f, 0.f, 0.f, 0.f, 0.f, 0.f};
      acc = __builtin_amdgcn_wmma_f32_16x16x4_f32(
          /*neg_a=*/false, A, /*neg_b=*/false, B,
          /*c_mod=*/(short)0, acc, /*reuse_a=*/false, /*reuse_b=*/false);
#pragma unroll
      for (int r = 0; r < 8; ++r) {
        int m = r + half * 8;
        float d = aam[r] + bb - 2.0f * acc[r];
        sd[m * SDS + st * 16 + l15] = d;
      }
    }
    __syncthreads();

    // ---- streaming top-32 selection: wave handles rows 2w, 2w+1 ----
#pragma unroll
    for (int rr = 0; rr < 2; ++rr) {
      const int m = wave * 2 + rr;
      unsigned long long cm = rr ? curmax1 : curmax0;
      for (int q = lane; q < PANEL; q += 32) {
        float d = sd[m * SDS + q];
        unsigned long long key =
            ((unsigned long long)fmono(d) << 32) | (unsigned int)(j0 + q);
        if (key < cm) {
          int e = KNBR - 1;
          while (e > 0) {
            unsigned long long prev = cand[(m * KNBR + (e - 1)) * 32 + lane];
            if (prev <= key) break;
            cand[(m * KNBR + e) * 32 + lane] = prev;
            --e;
          }
          cand[(m * KNBR + e) * 32 + lane] = key;
          cm = cand[(m * KNBR + (KNBR - 1)) * 32 + lane];
        }
      }
      if (rr) curmax1 = cm; else curmax0 = cm;
    }
    __syncthreads();
  }

  // ---- merge 32 sorted per-lane lists -> global top-32 per row ----
#pragma unroll
  for (int rr = 0; rr < 2; ++rr) {
    const int m = wave * 2 + rr;
    int pos = 0;
    for (int s = 0; s < KNBR; ++s) {
      unsigned long long head =
          (pos < KNBR) ? cand[(m * KNBR + pos) * 32 + lane] : ~0ull;
      unsigned long long mn = wave_min_u64(head);
      unsigned long long bal = __ballot(head == mn);
      int winner = __ffsll((unsigned long long)bal) - 1;
      if (lane == winner) pos++;
      if (lane == 0) midx[m * 32 + s] = (int)(mn & 0xFFFFFFFFu);
    }
  }

  // ---- covariance of 32 neighbors (one per lane) + min eigenvalue ----
#pragma unroll
  for (int rr = 0; rr < 2; ++rr) {
    const int m  = wave * 2 + rr;
    const int ni = midx[m * 32 + lane];
    float nx = X[ni], ny = Y[ni], nz = Z[ni];
    const float inv = 1.0f / (float)KNBR;
    float mx = wave_sum_f32(nx) * inv;
    float my = wave_sum_f32(ny) * inv;
    float mz = wave_sum_f32(nz) * inv;
    float ox = nx - mx, oy = ny - my, oz = nz - mz;
    float cxx = wave_sum_f32(ox * ox) * inv;
    float cxy = wave_sum_f32(ox * oy) * inv;
    float cxz = wave_sum_f32(ox * oz) * inv;
    float cyy = wave_sum_f32(oy * oy) * inv;
    float cyz = wave_sum_f32(oy * oz) * inv;
    float czz = wave_sum_f32(oz * oz) * inv;
    if (lane == 0)
      hout[cloud * NPTS + row0 + m] = mineig3(cxx, cxy, cxz, cyy, cyz, czz);
  }
}

// ---------------------------------------------------------------------------
// K2: bitonic sort of 8192 Harris values (descending value, ascending index)
// ---------------------------------------------------------------------------
extern "C" __global__ void __launch_bounds__(1024)
harris_sort(const float* __restrict__ h, float* __restrict__ sval,
            int* __restrict__ sidx) {
  extern __shared__ unsigned long long keys[];   // 8192 * 8B
  const int cloud = blockIdx.x;
  const float* hh = h + cloud * NPTS;

  for (int i = threadIdx.x; i < NPTS; i += 1024) {
    unsigned int b = ~fmono(hh[i]);  // descending order under unsigned compare
    keys[i] = ((unsigned long long)b << 32) | (unsigned int)i;
  }
  __syncthreads();

  for (int k = 2; k <= NPTS; k <<= 1) {
    for (int j = k >> 1; j > 0; j >>= 1) {
      for (int i = threadIdx.x; i < NPTS; i += 1024) {
        int ixj = i ^ j;
        if (ixj > i) {
          bool up = ((i & k) == 0);
          unsigned long long a = keys[i], b = keys[ixj];
          if ((a > b) == up) { keys[i] = b; keys[ixj] = a; }
        }
      }
      __syncthreads();
    }
  }

  for (int i = threadIdx.x; i < NTOPK; i += 1024) {
    int idx = (int)(keys[i] & 0xFFFFFFFFu);
    sidx[cloud * 1024 + i] = idx;
    sval[cloud * 1024 + i] = hh[idx];
  }
}

// ---------------------------------------------------------------------------
// K3: 819x819 scalar matching, top-15, final indices
// ---------------------------------------------------------------------------
extern "C" __global__ void __launch_bounds__(1024)
match_select(const float* __restrict__ sval, const int* __restrict__ sidx,
             float* __restrict__ out, int* __restrict__ it_ws) {
  __shared__ unsigned long long key[NTOPK];
  __shared__ float hs[NTOPK];
  __shared__ float ht[NTOPK];
  __shared__ int nn[NTOPK];
  __shared__ int sel[NSAMP];
  const int t = threadIdx.x;

  for (int i = t; i < NTOPK; i += 1024) {
    hs[i] = sval[i];
    ht[i] = sval[1024 + i];
  }
  __syncthreads();

  for (int j = t; j < NTOPK; j += 1024) {
    float hj = ht[j];
    float best = 3.4e38f;
    int   bi = 0;
    for (int i = 0; i < NTOPK; ++i) {
      float df = hs[i] - hj;
      float d  = df * df;
      if (d < best) { best = d; bi = i; }   // strict < keeps lowest i on ties
    }
    nn[j] = bi;
    key[j] = ((unsigned long long)fmono(best) << 32) | (unsigned int)j;
  }
  __syncthreads();

  for (int s = 0; s < NSAMP; ++s) {
    if (t < 32) {
      unsigned long long mn = ~0ull;
      for (int j = t; j < NTOPK; j += 32) {
        unsigned long long v = key[j];
        mn = (v < mn) ? v : mn;
      }
      mn = wave_min_u64(mn);
      if (t == 0) {
        int jw = (int)(mn & 0xFFFFFFFFu);
        sel[s] = jw;
        key[jw] = ~0ull;
      }
    }
    __syncthreads();
  }

  if (t < NSAMP) {
    int j  = sel[t];           // tgt_sel
    int i  = nn[j];            // src_sel
    int i0 = sidx[i];          // it0 = idx_s[src_sel]
    int i1 = sidx[1024 + j];   // it1 = idx_t[tgt_sel]
    it_ws[t]      = i0;
    it_ws[16 + t] = i1;
    out[15450 + t] = (float)i0;
    out[15465 + t] = (float)i1;
  }
}

// ---------------------------------------------------------------------------
// K4: gathers  (t_src | t_tgt | t_src_emb | t_tgt_emb)
// ---------------------------------------------------------------------------
extern "C" __global__ void __launch_bounds__(256)
gather_out(const float* __restrict__ src, const float* __restrict__ tgt,
           const float* __restrict__ semb, const float* __restrict__ temb,
           const int* __restrict__ it_ws, float* __restrict__ out) {
  const int i = blockIdx.x * blockDim.x + threadIdx.x;
  const int total = 45 + 45 + NCH * NSAMP * 2;
  if (i >= total) return;
  if (i < 45) {
    int c = i / NSAMP, s = i % NSAMP;
    out[i] = src[c * NPTS + it_ws[s]];
  } else if (i < 90) {
    int k = i - 45, c = k / NSAMP, s = k % NSAMP;
    out[i] = tgt[c * NPTS + it_ws[16 + s]];
  } else if (i < 90 + NCH * NSAMP) {
    int k = i - 90, c = k / NSAMP, s = k % NSAMP;
    out[i] = semb[c * NPTS + it_ws[s]];
  } else {
    int k = i - 90 - NCH * NSAMP, c = k / NSAMP, s = k % NSAMP;
    out[i] = temb[c * NPTS + it_ws[16 + s]];
  }
}

// ---------------------------------------------------------------------------
extern "C" void kernel_launch(void* const* d_in, const int* in_sizes, int n_in,
                              void* d_out, int out_size, void* d_ws, size_t ws_size,
                              hipStream_t stream) {
  const float* src  = (const float*)d_in[0];
  const float* tgt  = (const float*)d_in[1];
  const float* semb = (const float*)d_in[2];
  const float* temb = (const float*)d_in[3];
  float* out = (float*)d_out;

  char* ws = (char*)d_ws;
  float* h     = (float*)ws;                                  // 2*8192 f32
  float* sval  = (float*)(ws + 2 * NPTS * 4);                 // 2*1024 f32
  int*   sidx  = (int*)(ws + 2 * NPTS * 4 + 2 * 1024 * 4);    // 2*1024 i32
  int*   it_ws = (int*)(ws + 2 * NPTS * 4 + 4 * 1024 * 4);    // 32 i32

  // K1: 2 clouds * 512 row-tiles; dynamic LDS partitioned inside the kernel
  size_t smem1 = (size_t)16 * SDS * 4               // distance panel
               + (size_t)16 * KNBR * 32 * 8         // candidate lists
               + 16 * 32 * 4;                       // merged neighbor indices
  harris16x16x4<<<1024, 256, smem1, stream>>>(src, tgt, h);

  // K2: one block per cloud, 8192-key bitonic sort in LDS
  harris_sort<<<2, 1024, (size_t)NPTS * 8, stream>>>(h, sval, sidx);

  // K3: matching + top-15 + indices (also writes out[15450..15479])
  match_select<<<1, 1024, 0, stream>>>(sval, sidx, out, it_ws);

  // K4: gathers, out[0..15449]
  const int total = 45 + 45 + NCH * NSAMP * 2;
  gather_out<<<(total + 255) / 256, 256, 0, stream>>>(src, tgt, semb, temb,
                                                      it_ws, out);
}